// ReconModelMP_17549236371725
// MI455X (gfx1250) — compile-verified
//
#include <hip/hip_runtime.h>

typedef __attribute__((ext_vector_type(2))) float v2f;
typedef __attribute__((ext_vector_type(8))) float v8f;

#define IMG_S   1024
#define N_VERTS 468
#define N_TRI   256
#define NGROUP  16          // 256 triangles / 16 per WMMA
#define BATCH_PER_WAVE 8    // 8 waves * 8 batches * 16 px = 1024 px per row

// Rasterize silhouette mask via fp32 WMMA: edge functions are affine in the
// pixel -> W[16 tris x 16 px] = E[16x4] * P[4x16] with V_WMMA_F32_16X16X4_F32.
// Edge coefficients are orientation-normalized (signed area >= 0) so coverage
// is exactly  min(w0,w1,w2) >= 0,  and the OR over triangles is a max-reduce.
__global__ __launch_bounds__(256) void raster_wmma(
    const float* __restrict__ lm,    // [B][468][3]
    const int*   __restrict__ tri,   // [256][3]
    float*       __restrict__ out)   // [B][1024][1024]
{
    // Per (group, edge): 32 lanes x (a0, a1) A-fragment pairs.
    __shared__ v2f sA[NGROUP * 3 * 32];      // 12 KB

    const int t   = threadIdx.x;             // 0..255
    const int b   = blockIdx.y;              // image
    const int row = blockIdx.x;              // pixel row i

    // ---- per-workgroup coefficient setup: thread t handles triangle t ----
    {
        const int i0 = tri[t * 3 + 0];
        const int i1 = tri[t * 3 + 1];
        const int i2 = tri[t * 3 + 2];
        const float ax = lm[((size_t)b * N_VERTS + i0) * 3 + 0];
        const float ay = lm[((size_t)b * N_VERTS + i0) * 3 + 1];
        const float bx = lm[((size_t)b * N_VERTS + i1) * 3 + 0];
        const float by = lm[((size_t)b * N_VERTS + i1) * 3 + 1];
        const float cx = lm[((size_t)b * N_VERTS + i2) * 3 + 0];
        const float cy = lm[((size_t)b * N_VERTS + i2) * 3 + 1];
        // edge(p,q)(px,py) = -(qy-py)*px + (qx-px)*py + (px_*qy - py_*qx)
        float Ac[3], Bc[3], Cc[3];
        Ac[0] = -(cy - by); Bc[0] = (cx - bx); Cc[0] = bx * cy - by * cx;  // w0 = edge(b,c)
        Ac[1] = -(ay - cy); Bc[1] = (ax - cx); Cc[1] = cx * ay - cy * ax;  // w1 = edge(c,a)
        Ac[2] = -(by - ay); Bc[2] = (bx - ax); Cc[2] = ax * by - ay * bx;  // w2 = edge(a,b)
        // Orientation normalize: w0+w1+w2 == 2*signed_area (constant), so after
        // flipping negative-area triangles, "all w <= 0" is impossible (or all-zero,
        // which "all w >= 0" already accepts). Coverage == (min3(w) >= 0).
        const float area2 = (bx - ax) * (cy - ay) - (by - ay) * (cx - ax);
        const float s = (area2 < 0.0f) ? -1.0f : 1.0f;
        const int g = t >> 4, m = t & 15;
        #pragma unroll
        for (int e = 0; e < 3; ++e) {
            const int base = (g * 3 + e) * 32;
            v2f lohalf; lohalf[0] = s * Ac[e]; lohalf[1] = s * Bc[e];  // lanes 0-15: K=0,K=1
            v2f hihalf; hihalf[0] = s * Cc[e]; hihalf[1] = 0.0f;       // lanes 16-31: K=2,K=3
            sA[base + m]      = lohalf;
            sA[base + 16 + m] = hihalf;
        }
    }
    __syncthreads();

    const int  lane = t & 31;
    const int  wave = t >> 5;
    const bool lo   = lane < 16;

    // Hoist all 48 A-fragments into VGPRs (96 VGPRs, reused across batches).
    v2f afrag[NGROUP * 3];
    #pragma unroll
    for (int ge = 0; ge < NGROUP * 3; ++ge)
        afrag[ge] = sA[ge * 32 + lane];

    const float pyv  = 1.0f - ((float)row + 0.5f) * (1.0f / IMG_S);
    const v8f   zero = {};

    for (int k = 0; k < BATCH_PER_WAVE; ++k) {
        const int   j0 = (wave * BATCH_PER_WAVE + k) * 16;
        const float jj = (float)(j0 + (lane & 15));
        // B 4x16: K=0 row = px, K=1 = py (lanes 0-15); K=2 = 1, K=3 = 0 (lanes 16-31)
        v2f bfrag;
        bfrag[0] = lo ? (1.0f - (jj + 0.5f) * (1.0f / IMG_S)) : 1.0f;
        bfrag[1] = lo ? pyv : 0.0f;

        // covp[r] = max over processed triangles of min3(w0,w1,w2)
        float covp[8];
        #pragma unroll
        for (int r = 0; r < 8; ++r) covp[r] = -__builtin_inff();

        #pragma unroll
        for (int g = 0; g < NGROUP; ++g) {
            v8f d0 = __builtin_amdgcn_wmma_f32_16x16x4_f32(
                false, afrag[g * 3 + 0], false, bfrag, (short)0, zero, false, false);
            v8f d1 = __builtin_amdgcn_wmma_f32_16x16x4_f32(
                false, afrag[g * 3 + 1], false, bfrag, (short)0, zero, false, false);
            v8f d2 = __builtin_amdgcn_wmma_f32_16x16x4_f32(
                false, afrag[g * 3 + 2], false, bfrag, (short)0, zero, false, false);
            #pragma unroll
            for (int r = 0; r < 8; ++r) {
                const float mn = fminf(fminf(d0[r], d1[r]), d2[r]);   // -> v_min3_num_f32
                covp[r] = fmaxf(covp[r], mn);
            }
        }
        // Reduce the 8 row-slots, then merge M=0..7 (lanes 0-15) with M=8..15
        // (lanes 16-31): both halves hold the same pixel N = lane%16.
        float m0 = fmaxf(fmaxf(covp[0], covp[1]), covp[2]);
        float m1 = fmaxf(fmaxf(covp[3], covp[4]), covp[5]);
        float m2 = fmaxf(covp[6], covp[7]);
        float mm = fmaxf(fmaxf(m0, m1), m2);
        mm = fmaxf(mm, __shfl_xor(mm, 16, 32));
        if (lo)
            out[((size_t)b * IMG_S + row) * IMG_S + j0 + lane] = (mm >= 0.0f) ? 1.0f : 0.0f;
    }
}

extern "C" void kernel_launch(void* const* d_in, const int* in_sizes, int n_in,
                              void* d_out, int out_size, void* d_ws, size_t ws_size,
                              hipStream_t stream) {
    const float* lm  = (const float*)d_in[0];   // landmarks [B][468][3] fp32
    const int*   tri = (const int*)d_in[1];     // tri [256][3] int32
    float*       out = (float*)d_out;           // mask [B][1024][1024] as 0/1 float

    const int B = in_sizes[0] / (N_VERTS * 3);  // = 8
    dim3 grid(IMG_S, B);
    raster_wmma<<<grid, 256, 0, stream>>>(lm, tri, out);
}